// MambaLayer_2662879724368
// MI455X (gfx1250) — compile-verified
//
#include <hip/hip_runtime.h>
#include <hip/hip_bf16.h>

typedef __attribute__((ext_vector_type(2))) float v2f;
typedef __attribute__((ext_vector_type(8))) float v8f;

#define BATCH   4
#define SEQL    2048
#define DMODEL  1024
#define DINNER  2048
#define DSTATE  16
#define DTRANK  64
#define DCONV   4
#define MTOT    (BATCH * SEQL)        /* 8192 rows */
#define XZLD    (2 * DINNER)          /* 4096 */
#define XDBLLD  (DTRANK + 2 * DSTATE) /* 96 */

// ---------------------------------------------------------------------------
// fp32 WMMA GEMM: C = epilogue(A @ W^T), all row-major.
// Wave tile = (16*MI) x (16*JT). Block = 256 threads = 8 waves stacked in M
// -> block tile (128*MI) x (16*JT). K chunked by 16 (4 k-steps of 4):
// all fragments of a chunk are loaded first, then 4*MI*JT WMMAs issue
// back-to-back. All loop bounds compile-time -> no acc copies, batched loads.
// EPI 0: plain store; EPI 1: softplus(acc + bias[col]).
// ---------------------------------------------------------------------------
template <int MI, int JT, int EPI>
__global__ __launch_bounds__(256) void wmma_gemm_f32(
    const float* __restrict__ A, const float* __restrict__ W,
    float* __restrict__ C, int K, int lda, int ldb, int ldc,
    const float* __restrict__ bias)
{
    const int wave = threadIdx.x >> 5;
    const int lane = threadIdx.x & 31;
    const int half = lane >> 4;        // 0: lanes 0-15, 1: lanes 16-31
    const int r    = lane & 15;

    const int m0 = (blockIdx.y * 8 + wave) * (16 * MI);
    const int n0 = blockIdx.x * (16 * JT);

    const float* Ap[MI];
#pragma unroll
    for (int mi = 0; mi < MI; ++mi)
        Ap[mi] = A + (size_t)(m0 + 16 * mi + r) * lda + 2 * half;

    const float* Wp[JT];
#pragma unroll
    for (int j = 0; j < JT; ++j)
        Wp[j] = W + (size_t)(n0 + 16 * j + r) * ldb + 2 * half;

    v8f acc[MI][JT];
#pragma unroll
    for (int mi = 0; mi < MI; ++mi)
#pragma unroll
        for (int j = 0; j < JT; ++j) acc[mi][j] = 0.f;

    for (int k0 = 0; k0 < K; k0 += 16) {
        // ---- load all fragments of this K-chunk (batched vmem) ----
        v2f afr[MI][4];
#pragma unroll
        for (int mi = 0; mi < MI; ++mi)
#pragma unroll
            for (int kk = 0; kk < 4; ++kk)
                afr[mi][kk] = *(const v2f*)(Ap[mi] + k0 + 4 * kk);

        v2f bfr[JT][4];
#pragma unroll
        for (int j = 0; j < JT; ++j)
#pragma unroll
            for (int kk = 0; kk < 4; ++kk)
                bfr[j][kk] = *(const v2f*)(Wp[j] + k0 + 4 * kk);

        // ---- 4*MI*JT WMMAs back-to-back ----
#pragma unroll
        for (int kk = 0; kk < 4; ++kk)
#pragma unroll
            for (int mi = 0; mi < MI; ++mi)
#pragma unroll
                for (int j = 0; j < JT; ++j)
                    acc[mi][j] = __builtin_amdgcn_wmma_f32_16x16x4_f32(
                        false, afr[mi][kk], false, bfr[j][kk],
                        (short)0, acc[mi][j], false, false);
    }

    // ---- store: acc[mi][j][i] -> C[m0+16*mi+8*half+i, n0+16*j+r] ----
#pragma unroll
    for (int mi = 0; mi < MI; ++mi) {
#pragma unroll
        for (int j = 0; j < JT; ++j) {
            const int col = n0 + 16 * j + r;
            float bj = 0.f;
            if (EPI == 1) bj = bias[col];
#pragma unroll
            for (int i = 0; i < 8; ++i) {
                const int row = m0 + 16 * mi + 8 * half + i;
                float v = acc[mi][j][i];
                if (EPI == 1) {
                    v += bj;
                    v = (v > 20.f) ? v : log1pf(expf(v));  // softplus
                }
                C[(size_t)row * ldc + col] = v;
            }
        }
    }
}

// ---------------------------------------------------------------------------
// Depthwise causal conv1d (K=4) + bias + SiLU:
//   xs[b,l,d] = silu( sum_k cw[d,k]*xr[b,l-3+k,d] + cb[d] )
// xr lives in columns [0, DINNER) of xz (row stride XZLD).
// ---------------------------------------------------------------------------
__global__ __launch_bounds__(256) void conv_silu_kernel(
    const float* __restrict__ xz, const float* __restrict__ cw,
    const float* __restrict__ cb, float* __restrict__ xs)
{
    const size_t idx = (size_t)blockIdx.x * 256 + threadIdx.x; // MTOT*DINNER
    const int    d   = (int)(idx % DINNER);
    const size_t t   = idx / DINNER;       // b*SEQL + l
    const int    l   = (int)(t % SEQL);
    const size_t b   = t / SEQL;

    float acc = cb[d];
#pragma unroll
    for (int k = 0; k < DCONV; ++k) {
        const int ll = l - (DCONV - 1) + k;
        if (ll >= 0)
            acc += cw[d * DCONV + k] * xz[(b * SEQL + (size_t)ll) * XZLD + d];
    }
    const float sig = 1.f / (1.f + expf(-acc));
    xs[idx] = acc * sig;
}

// ---------------------------------------------------------------------------
// Selective scan fused with output gating:
//   h[s] <- exp(dt*a_diag[s])*h[s] + (dt*x)*Bsel[s]
//   y     = sum_s h[s]*Csel[s] + D[d]*x ;  yz = y * silu(z)
// One thread per (b,d) channel; 16-state h in registers; per-(b,l) Bsel/Csel
// (32 floats) staged in LDS and shared by the 256 channels of the block.
// ---------------------------------------------------------------------------
__global__ __launch_bounds__(256) void scan_gate_kernel(
    const float* __restrict__ dtb, const float* __restrict__ xs,
    const float* __restrict__ xdbl, const float* __restrict__ xz,
    const float* __restrict__ A_log, const float* __restrict__ Dvec,
    float* __restrict__ yz)
{
    const int b = blockIdx.y;
    const int d = blockIdx.x * 256 + threadIdx.x;

    __shared__ float sB[DSTATE];
    __shared__ float sC[DSTATE];

    float adiag[DSTATE];
#pragma unroll
    for (int s = 0; s < DSTATE; ++s) adiag[s] = -expf(A_log[s]);  // broadcast

    float h[DSTATE];
#pragma unroll
    for (int s = 0; s < DSTATE; ++s) h[s] = 0.f;

    const float Dd = Dvec[d];

    for (int l = 0; l < SEQL; ++l) {
        const size_t t = (size_t)b * SEQL + l;
        if (threadIdx.x < 2 * DSTATE) {
            const float v = xdbl[t * XDBLLD + DTRANK + threadIdx.x];
            if (threadIdx.x < DSTATE) sB[threadIdx.x] = v;
            else                      sC[threadIdx.x - DSTATE] = v;
        }
        __syncthreads();

        const float dtv = dtb[t * DINNER + d];
        const float xv  = xs[t * DINNER + d];
        const float zv  = xz[t * XZLD + DINNER + d];
        const float du  = dtv * xv;

        float y = 0.f;
#pragma unroll
        for (int s = 0; s < DSTATE; ++s) {
            const float ab = expf(dtv * adiag[s]);
            h[s] = ab * h[s] + du * sB[s];
            y += h[s] * sC[s];
        }
        y += Dd * xv;

        const float sig = 1.f / (1.f + expf(-zv));
        yz[t * DINNER + d] = y * (zv * sig);
        __syncthreads();
    }
}

// ---------------------------------------------------------------------------
// Launch
// ---------------------------------------------------------------------------
extern "C" void kernel_launch(void* const* d_in, const int* in_sizes, int n_in,
                              void* d_out, int out_size, void* d_ws, size_t ws_size,
                              hipStream_t stream)
{
    const float* x     = (const float*)d_in[0];
    const float* W_in  = (const float*)d_in[1];
    const float* convw = (const float*)d_in[2];
    const float* convb = (const float*)d_in[3];
    const float* A_log = (const float*)d_in[4];
    const float* Dvec  = (const float*)d_in[5];
    const float* W_x   = (const float*)d_in[6];
    const float* W_dt  = (const float*)d_in[7];
    const float* b_dt  = (const float*)d_in[8];
    const float* W_out = (const float*)d_in[9];
    float* out = (float*)d_out;

    float* ws   = (float*)d_ws;
    float* xz   = ws;                                 // MTOT*4096
    float* xs   = xz   + (size_t)MTOT * XZLD;         // MTOT*2048
    float* xdbl = xs   + (size_t)MTOT * DINNER;       // MTOT*96
    float* dtb  = xdbl + (size_t)MTOT * XDBLLD;       // MTOT*2048
    float* yz   = dtb  + (size_t)MTOT * DINNER;       // MTOT*2048

    dim3 blk(256);

    // 1) xz = x @ W_in^T   (8192 x 4096, K=1024); block tile 256x64
    wmma_gemm_f32<2, 4, 0><<<dim3(XZLD / 64, MTOT / 256), blk, 0, stream>>>(
        x, W_in, xz, DMODEL, DMODEL, DMODEL, XZLD, nullptr);

    // 2) depthwise causal conv + SiLU -> xs
    conv_silu_kernel<<<(MTOT * DINNER) / 256, blk, 0, stream>>>(xz, convw, convb, xs);

    // 3) x_dbl = xs @ W_x^T  (8192 x 96, K=2048); one column-block of 96
    wmma_gemm_f32<2, 6, 0><<<dim3(1, MTOT / 256), blk, 0, stream>>>(
        xs, W_x, xdbl, DINNER, DINNER, DINNER, XDBLLD, nullptr);

    // 4) dt = softplus(dt_r @ W_dt^T + b_dt)  (8192 x 2048, K=64)
    wmma_gemm_f32<2, 4, 1><<<dim3(DINNER / 64, MTOT / 256), blk, 0, stream>>>(
        xdbl, W_dt, dtb, DTRANK, XDBLLD, DTRANK, DINNER, b_dt);

    // 5) selective scan + gating -> yz
    scan_gate_kernel<<<dim3(DINNER / 256, BATCH), blk, 0, stream>>>(
        dtb, xs, xdbl, xz, A_log, Dvec, yz);

    // 6) out = yz @ W_out^T  (8192 x 1024, K=2048)
    wmma_gemm_f32<2, 4, 0><<<dim3(DMODEL / 64, MTOT / 256), blk, 0, stream>>>(
        yz, W_out, out, DINNER, DINNER, DINNER, DMODEL, nullptr);
}